// ElasticPooling_85718957293684
// MI455X (gfx1250) — compile-verified
//
#include <hip/hip_runtime.h>

// out[b,p,w,c] = max_{r<16, j<2} x[b, c, 16p+r, 2w+j]
//   x:   (64, 8, 2048, 32) f32   (128 MiB)
//   out: (64, 128, 16, 8)  f32   (4 MiB)
// Pure HBM-bound streaming reduction; uses CDNA5 async global->LDS path
// (global_load_async_to_lds_b128 + s_wait_asynccnt) with double buffering.

#define NB 4  // (b,p) blocks per workgroup

__global__ __launch_bounds__(256)
void ElasticPooling_85718957293684_kernel(const float* __restrict__ x,
                                          float* __restrict__ out) {
  const int lane = threadIdx.x & 31;
  const int wv   = threadIdx.x >> 5;   // wave id == channel c (0..7)
  const int q0   = blockIdx.x * NB;    // first block index; q = b*128 + p

  __shared__ __align__(16) float lds[2][8][512];  // 2 bufs x 8 waves x 2KiB

  // LDS byte offsets for this lane in each buffer (low 32 bits of the
  // generic pointer are the LDS offset on gfx1250).
  unsigned lds_off0 =
      (unsigned)(unsigned long long)(&lds[0][wv][0]) + (unsigned)lane * 16u;
  unsigned lds_off1 =
      (unsigned)(unsigned long long)(&lds[1][wv][0]) + (unsigned)lane * 16u;

  // Global tile base for block q: contiguous 512 floats (16 rows x 32 w).
  auto tile_base = [&](int q) -> const float* {
    int b = q >> 7, p = q & 127;
    return x + ((size_t)(b * 8 + wv) * 2048u + (size_t)p * 16u) * 32u;
  };

  // Issue the 4 async 128b-per-lane loads covering one 2 KiB tile.
#define ISSUE(q, ldsoff)                                                     \
  do {                                                                       \
    const float* gp = tile_base(q) + lane * 4;                               \
    asm volatile(                                                            \
        "global_load_async_to_lds_b128 %0, %1, off\n\t"                      \
        "global_load_async_to_lds_b128 %0, %1, off offset:512\n\t"           \
        "global_load_async_to_lds_b128 %0, %1, off offset:1024\n\t"          \
        "global_load_async_to_lds_b128 %0, %1, off offset:1536"              \
        :: "v"(ldsoff), "v"(gp)                                              \
        : "memory");                                                         \
  } while (0)

#define WAIT_ASYNC_4() asm volatile("s_wait_asynccnt 4" ::: "memory")
#define WAIT_ASYNC_0() asm volatile("s_wait_asynccnt 0" ::: "memory")

  ISSUE(q0, lds_off0);

  for (int i = 0; i < NB; ++i) {
    if (i + 1 < NB) {
      ISSUE(q0 + i + 1, ((i + 1) & 1) ? lds_off1 : lds_off0);
      WAIT_ASYNC_4();   // first 4 outstanding (tile i) are done; tile i+1 in flight
    } else {
      WAIT_ASYNC_0();
    }

    // Reduce tile i from LDS. Element e = pass*32 + lane (float4 units):
    //   row r = 4*pass + (lane>>3), w-pair = lane&7.
    const float4* src = (const float4*)(&lds[i & 1][wv][0]) + lane;
    float acc0 = -__builtin_inff();
    float acc1 = -__builtin_inff();
#pragma unroll
    for (int pass = 0; pass < 4; ++pass) {
      float4 v = src[pass * 32];
      acc0 = fmaxf(acc0, fmaxf(v.x, v.y));  // w = 2*(lane&7)
      acc1 = fmaxf(acc1, fmaxf(v.z, v.w));  // w = 2*(lane&7)+1
    }
    // Max across the 4 lanes sharing the same (lane&7): xor over bits 3,4.
    acc0 = fmaxf(acc0, __shfl_xor(acc0, 8, 32));
    acc1 = fmaxf(acc1, __shfl_xor(acc1, 8, 32));
    acc0 = fmaxf(acc0, __shfl_xor(acc0, 16, 32));
    acc1 = fmaxf(acc1, __shfl_xor(acc1, 16, 32));

    if (lane < 8) {
      int q = q0 + i;  // q = b*128 + p
      // out flat index: (q*16 + w)*8 + c
      float* o = out + ((size_t)q * 16u + 2u * (unsigned)lane) * 8u + wv;
      o[0] = acc0;
      o[8] = acc1;
    }
  }
#undef ISSUE
#undef WAIT_ASYNC_4
#undef WAIT_ASYNC_0
}

extern "C" void kernel_launch(void* const* d_in, const int* in_sizes, int n_in,
                              void* d_out, int out_size, void* d_ws, size_t ws_size,
                              hipStream_t stream) {
  (void)in_sizes; (void)n_in; (void)out_size; (void)d_ws; (void)ws_size;
  const float* x = (const float*)d_in[0];
  float* out = (float*)d_out;
  // 64*128 = 8192 (b,p) blocks, NB per workgroup.
  dim3 grid(8192 / NB), block(256);
  ElasticPooling_85718957293684_kernel<<<grid, block, 0, stream>>>(x, out);
}